// EventPropLinear_73452530696742
// MI455X (gfx1250) — compile-verified
//
#include <hip/hip_runtime.h>

// ---------------------------------------------------------------------------
// EventPropLinear for MI455X (gfx1250, wave32, WMMA)
//   Pass 1: cur[row, o] = sum_k X[row, k] * W[o, k]   (row = t*B + b, t<127)
//           fp32 GEMM using V_WMMA_F32_16X16X4_F32, LDS-staged tiles.
//   Pass 2: sequential LIF scan over t per (b, o), coalesced over o.
// ---------------------------------------------------------------------------

typedef float v2f __attribute__((ext_vector_type(2)));
typedef float v8f __attribute__((ext_vector_type(8)));

#define T_STEPS   128
#define BATCH     64
#define IN_DIM    1024
#define OUT_DIM   1024
#define M_TOTAL   ((T_STEPS - 1) * BATCH)   // 8128

#define BM 32          // block tile rows (M)
#define BN 256         // block tile cols (N)
#define KC 32          // K chunk staged in LDS
#define KPAD (KC + 4)  // 36 floats: conflict-free frag reads, float4-alignable

// membrane constants (DT=1, TAU_M=10, TAU_S=1)
#define AM_C 0.9f      // 1 - DT/TAU_M
#define BM_C 0.1f      // DT/TAU_M
#define AS_C 0.0f      // 1 - DT/TAU_S

__global__ void __launch_bounds__(256)
eventprop_gemm_f32_wmma(const float* __restrict__ X,   // [M_TOTAL, IN_DIM]
                        const float* __restrict__ W,   // [OUT_DIM, IN_DIM]
                        float* __restrict__ cur)       // [M_TOTAL, OUT_DIM]
{
    __shared__ float As[BM][KPAD];   //  4.5 KB
    __shared__ float Bs[BN][KPAD];   // 36.0 KB

    const int tid   = threadIdx.x;          // 0..255
    const int lane  = tid & 31;             // wave32
    const int wave  = tid >> 5;             // 0..7
    const int mTile = wave & 1;             // 0..1  -> 16-row tile in block
    const int nGrp  = wave >> 1;            // 0..3  -> 64-col group in block

    const int blockM = blockIdx.y * BM;     // 254 blocks
    const int blockN = blockIdx.x * BN;     // 4 blocks

    // WMMA fp32 16x16x4 fragment addressing (per CDNA5 ISA 32-bit layouts):
    //   A: lanes 0-15 -> M=lane, VGPR0=K0 VGPR1=K1; lanes 16-31 -> K2/K3
    //   B: mirrored (N=lane&15, same K split); C/D: VGPR v -> M=v (+8 hi half)
    const int fm  = lane & 15;              // A row within tile / B col within tile
    const int kb  = (lane >> 4) << 1;       // 0 or 2
    const int hi8 = (lane >> 4) << 3;       // 0 or 8 (C/D upper-half rows)

    v8f acc[4] = {};                        // 16x64 per wave: 4 N-tiles

    for (int kc = 0; kc < IN_DIM; kc += KC) {
        // ---- stage A: 32 rows x 32 k, one float4 per thread (coalesced) ----
        {
            const int r = tid >> 3;                 // 0..31
            const int k = (tid & 7) << 2;           // 0,4,...,28
            const float4 v = *(const float4*)(X + (blockM + r) * IN_DIM + kc + k);
            *(float4*)&As[r][k] = v;
        }
        // ---- stage B: 256 rows x 32 k, eight float4 per thread (coalesced) ----
        {
            const int rb = tid >> 3;                // 0..31
            const int k  = (tid & 7) << 2;
#pragma unroll
            for (int i = 0; i < 8; ++i) {
                const int r = rb + (i << 5);        // 0..255
                const float4 v = *(const float4*)(W + (blockN + r) * IN_DIM + kc + k);
                *(float4*)&Bs[r][k] = v;
            }
        }
        __syncthreads();

        // ---- 8 WMMA k-steps over the staged chunk ----
#pragma unroll
        for (int k0 = 0; k0 < KC; k0 += 4) {
            v2f a = *(const v2f*)&As[mTile * 16 + fm][k0 + kb];   // ds_load_b64
#pragma unroll
            for (int j = 0; j < 4; ++j) {
                const int oc = nGrp * 64 + j * 16 + fm;
                v2f b = *(const v2f*)&Bs[oc][k0 + kb];            // ds_load_b64
                acc[j] = __builtin_amdgcn_wmma_f32_16x16x4_f32(
                    /*neg_a=*/false, a, /*neg_b=*/false, b,
                    /*c_mod=*/(short)0, acc[j],
                    /*reuse_a=*/false, /*reuse_b=*/false);
            }
        }
        __syncthreads();
    }

    // ---- write back: D VGPR v holds row (v + hi8), col = lane&15 ----
    const int rowBase = blockM + mTile * 16 + hi8;
    const int col     = fm;
#pragma unroll
    for (int j = 0; j < 4; ++j) {
        const int oc = blockN + nGrp * 64 + j * 16 + col;
#pragma unroll
        for (int v = 0; v < 8; ++v) {
            cur[(size_t)(rowBase + v) * OUT_DIM + oc] = acc[j][v];
        }
    }
}

__global__ void __launch_bounds__(256)
eventprop_scan(const float* __restrict__ cur,   // [T-1, B, OUT]
               float* __restrict__ out)         // [T,   B, OUT]
{
    const int idx = blockIdx.x * blockDim.x + threadIdx.x;  // (b,o) flat, o fastest
    const int BO  = BATCH * OUT_DIM;
    if (idx >= BO) return;

    float V = 0.0f, I = 0.0f;
    out[idx] = 0.0f;                                        // t = 0 row is zero
    for (int t = 1; t < T_STEPS; ++t) {
        const float c  = cur[(size_t)(t - 1) * BO + idx];
        const float Vn = AM_C * V + BM_C * I;
        const float In = AS_C * I + c;
        const bool  sp = (Vn > 1.0f);
        out[(size_t)t * BO + idx] = sp ? 1.0f : 0.0f;
        V = sp ? 0.0f : Vn;
        I = In;
    }
}

extern "C" void kernel_launch(void* const* d_in, const int* in_sizes, int n_in,
                              void* d_out, int out_size, void* d_ws, size_t ws_size,
                              hipStream_t stream) {
    const float* x = (const float*)d_in[0];   // [T, B, IN]; rows 0..T-2 used
    const float* w = (const float*)d_in[1];   // [OUT, IN]
    float* out = (float*)d_out;               // [T, B, OUT]
    float* cur = (float*)d_ws;                // [T-1, B, OUT] = ~31.8 MB scratch

    dim3 gGemm(OUT_DIM / BN, M_TOTAL / BM);   // (4, 254)
    eventprop_gemm_f32_wmma<<<gGemm, 256, 0, stream>>>(x, w, cur);

    const int BO = BATCH * OUT_DIM;           // 65536
    eventprop_scan<<<BO / 256, 256, 0, stream>>>(cur, out);
}